// Classifier_42588895707508
// MI455X (gfx1250) — compile-verified
//
#include <hip/hip_runtime.h>

// ---------------------------------------------------------------------------
// Problem constants (from the reference): B=512, S=256, H=768, C=2, K=2H=1536
// ---------------------------------------------------------------------------
#define BB 512
#define SS 256
#define HH 768
#define CC 2
#define KK 1536

typedef __attribute__((ext_vector_type(2))) float v2f;
typedef __attribute__((ext_vector_type(8))) float v8f;

// ---------------------------------------------------------------------------
// Kernel 1: masked max pooling.
// grid = 1024 (tensor * 512 batches), block = 192 threads (6 waves).
// Each thread owns 4 consecutive channels -> float4 loads, 3072B/row coalesced.
// eff = index of first zero in mask row; 0 or "no zero" -> full S (argmin rule).
// ---------------------------------------------------------------------------
__global__ __launch_bounds__(192) void pool_kernel(const float* __restrict__ x1,
                                                   const float* __restrict__ x2,
                                                   const int* __restrict__ m1,
                                                   const int* __restrict__ m2,
                                                   float* __restrict__ P) {
  const int which = blockIdx.x >> 9;     // 0 -> x1/m1, 1 -> x2/m2
  const int b     = blockIdx.x & 511;
  const float* __restrict__ x  = which ? x2 : x1;
  const int*   __restrict__ mk = which ? m2 : m1;
  const int t = threadIdx.x;

  // ---- first-zero index of mask row (argmin over int32 prefix-mask) ----
  __shared__ int red[192];
  int v = SS;
  for (int i = t; i < SS; i += 192) {
    if (mk[(size_t)b * SS + i] == 0) v = min(v, i);
  }
  red[t] = v;
  __syncthreads();
  for (int off = 96; off > 0; off >>= 1) {
    if (t < off) red[t] = min(red[t], red[t + off]);
    __syncthreads();
  }
  int eff = red[0];
  if (eff == 0) eff = SS;   // argmin==0 (mask[0]==0) or no zero -> full length

  // ---- streaming max over valid timesteps, float4 per thread ----
  const float4* __restrict__ xr =
      (const float4*)(x + (size_t)b * SS * HH);   // [S][192] float4
  float ninf = -__builtin_inff();
  float4 mx = make_float4(ninf, ninf, ninf, ninf);
  for (int s = 0; s < eff; ++s) {
    float4 val = xr[(size_t)s * (HH / 4) + t];
    mx.x = fmaxf(mx.x, val.x);
    mx.y = fmaxf(mx.y, val.y);
    mx.z = fmaxf(mx.z, val.z);
    mx.w = fmaxf(mx.w, val.w);
  }

  float4* __restrict__ pout =
      (float4*)(P + (size_t)b * KK + (size_t)which * HH);
  pout[t] = mx;
}

// ---------------------------------------------------------------------------
// Kernel 2: pad W[2][1536] -> Wp[16][1536] (rows >= 2 zero) so WMMA B-fragment
// loads are uniform (no divergence before v_wmma; EXEC stays all-1s).
// ---------------------------------------------------------------------------
__global__ __launch_bounds__(256) void wpad_kernel(const float* __restrict__ W,
                                                   float* __restrict__ Wp) {
  int i = blockIdx.x * 256 + threadIdx.x;   // 16*1536 = 24576 total
  int n = i / KK;
  Wp[i] = (n < CC) ? W[i] : 0.0f;
}

// ---------------------------------------------------------------------------
// Kernel 3: OUT[512][2] = P[512][1536] @ Wp^T + bias, via V_WMMA_F32_16X16X4_F32.
// One wave (32 threads) per 16-batch tile; K looped in steps of 4.
//
// A (16x4 f32) fragment layout per ISA 7.12.2:
//   lanes 0-15 : M=lane,    VGPR0=K0, VGPR1=K1
//   lanes 16-31: M=lane-16, VGPR0=K2, VGPR1=K3
// B (4x16) mirrors: lane gives N=lane&15, VGPR r holds K = r + 2*(lane>>4).
// C/D (16x16 f32): VGPR r, lanes 0-15 -> M=r, N=lane; lanes 16-31 -> M=r+8.
// ---------------------------------------------------------------------------
__global__ __launch_bounds__(32) void gemm_wmma(const float* __restrict__ P,
                                                const float* __restrict__ Wp,
                                                const float* __restrict__ bias,
                                                float* __restrict__ out) {
  const int lane  = threadIdx.x;        // 0..31
  const int m16   = lane & 15;
  const int hi    = lane >> 4;          // 0 or 1
  const int kbase = hi * 2;             // hi lanes carry K=2,3 of each step
  const int arow  = blockIdx.x * 16 + m16;

  const float* __restrict__ prow = P + (size_t)arow * KK;
  const float* __restrict__ wrow = Wp + (size_t)m16 * KK;   // N = m16 (padded)

  v8f acc = {};
  for (int k0 = 0; k0 < KK; k0 += 4) {
    v2f a, bfrag;
    a.x     = prow[k0 + kbase];
    a.y     = prow[k0 + kbase + 1];
    bfrag.x = wrow[k0 + kbase];
    bfrag.y = wrow[k0 + kbase + 1];
    // 8 args: (neg_a, A, neg_b, B, c_mod, C, reuse_a, reuse_b)
    acc = __builtin_amdgcn_wmma_f32_16x16x4_f32(
        false, a, false, bfrag, (short)0, acc, false, false);
  }

  // Only output channels 0..1 are real; lanes 0,1,16,17 hold them.
  if (m16 < CC) {
    const float bv = bias[m16];
    #pragma unroll
    for (int r = 0; r < 8; ++r) {
      int brow = blockIdx.x * 16 + r + 8 * hi;   // D row for VGPR r
      out[(size_t)brow * CC + m16] = acc[r] + bv;
    }
  }
}

// ---------------------------------------------------------------------------
// Launcher. Inputs (setup_inputs order): x1, x2, m1, m2, W, b.
// Workspace layout: P (512*1536 f32 = 3 MB) | Wp (16*1536 f32 = 96 KB).
// ---------------------------------------------------------------------------
extern "C" void kernel_launch(void* const* d_in, const int* in_sizes, int n_in,
                              void* d_out, int out_size, void* d_ws, size_t ws_size,
                              hipStream_t stream) {
  (void)in_sizes; (void)n_in; (void)out_size; (void)ws_size;

  const float* x1   = (const float*)d_in[0];
  const float* x2   = (const float*)d_in[1];
  const int*   m1   = (const int*)d_in[2];
  const int*   m2   = (const int*)d_in[3];
  const float* W    = (const float*)d_in[4];
  const float* bias = (const float*)d_in[5];
  float*       out  = (float*)d_out;

  float* P  = (float*)d_ws;
  float* Wp = P + (size_t)BB * KK;

  pool_kernel<<<2 * BB, 192, 0, stream>>>(x1, x2, m1, m2, P);
  wpad_kernel<<<(16 * KK) / 256, 256, 0, stream>>>(W, Wp);
  gemm_wmma<<<BB / 16, 32, 0, stream>>>(P, Wp, bias, out);
}